// LengthRegulator_53987738910726
// MI455X (gfx1250) — compile-verified
//
#include <hip/hip_runtime.h>
#include <hip/hip_bf16.h>

// Problem constants from the reference
#define BATCH      32
#define T_PH       1024
#define DDIM       384          // floats per row -> 1536 bytes
#define MAX_FRAMES 4096
#define ROW_BYTES  (DDIM * 4)   // 1536
#define WAVES_PER_BLOCK 8

// ---------------------------------------------------------------------------
// Kernel 1: per-batch inclusive cumsum of clamped durations (1024 ints/row).
// 32 blocks x 256 threads, each thread handles 4 elements via int4.
// ---------------------------------------------------------------------------
__global__ __launch_bounds__(256) void lr_scan_kernel(const int* __restrict__ dur,
                                                      int* __restrict__ cum) {
    __shared__ int part[256];
    const int b = blockIdx.x;
    const int t = threadIdx.x;

    int4 d = ((const int4*)(dur + (size_t)b * T_PH))[t];
    d.x = max(d.x, 1); d.y = max(d.y, 1); d.z = max(d.z, 1); d.w = max(d.w, 1);
    const int s0 = d.x;
    const int s1 = s0 + d.y;
    const int s2 = s1 + d.z;
    const int s3 = s2 + d.w;

    part[t] = s3;
    __syncthreads();
    // Hillis-Steele inclusive scan over 256 partials
    for (int off = 1; off < 256; off <<= 1) {
        int v = (t >= off) ? part[t - off] : 0;
        __syncthreads();
        part[t] += v;
        __syncthreads();
    }
    const int prefix = (t > 0) ? part[t - 1] : 0;

    int4 c;
    c.x = prefix + s0; c.y = prefix + s1; c.z = prefix + s2; c.w = prefix + s3;
    ((int4*)(cum + (size_t)b * T_PH))[t] = c;
}

// ---------------------------------------------------------------------------
// Kernel 2: gather. One wave (32 lanes) per output frame.
//  - block loads cum[b,:] into LDS (coalesced int4)
//  - wave-uniform binary search (searchsorted side='right')
//  - row copy via CDNA5 async global->LDS->global DMA (ASYNCcnt-tracked)
// ---------------------------------------------------------------------------
__global__ __launch_bounds__(256) void lr_gather_kernel(const float* __restrict__ x,
                                                        const int* __restrict__ cum,
                                                        float* __restrict__ out) {
    __shared__ int cum_s[T_PH];
    __shared__ __align__(16) char stage[WAVES_PER_BLOCK * ROW_BYTES];

    const int b    = blockIdx.y;
    const int tid  = threadIdx.x;
    const int w    = tid >> 5;   // wave id within block: 0..7
    const int lane = tid & 31;

    // Stage the cumulative-sum row into LDS (256 threads x int4 = 4KB)
    ((int4*)cum_s)[tid] = ((const int4*)(cum + (size_t)b * T_PH))[tid];
    __syncthreads();

    const int f     = blockIdx.x * WAVES_PER_BLOCK + w;   // output frame (wave-uniform)
    const int total = cum_s[T_PH - 1];

    float* orow = out + ((size_t)b * MAX_FRAMES + (size_t)f) * DDIM;

    if (f >= total) {
        // Masked tail: write zeros (3 x float4 per lane = 1536B per wave)
        float4 z; z.x = 0.f; z.y = 0.f; z.z = 0.f; z.w = 0.f;
        float4* o4 = (float4*)orow;
        o4[lane]      = z;
        o4[lane + 32] = z;
        o4[lane + 64] = z;
        return;
    }

    // searchsorted(cum, f, side='right'): first index with cum[idx] > f.
    // Lane-uniform -> LDS broadcast reads, 10 iterations.
    int lo = 0, hi = T_PH;
    #pragma unroll
    for (int it = 0; it < 10; ++it) {
        int mid = (lo + hi) >> 1;
        if (cum_s[mid] <= f) lo = mid + 1; else hi = mid;
    }
    const int row = lo;   // f < total guarantees row <= T_PH-1

    const float* srow = x + ((size_t)b * T_PH + (size_t)row) * DDIM;

    // Per-lane addresses: 16 bytes per lane per transfer, 512B per instruction.
    // Low 32 bits of a generic LDS pointer are the workgroup-relative LDS offset.
    unsigned lds_a = (unsigned)(size_t)(&stage[w * ROW_BYTES + lane * 16]);
    unsigned long long ga = (unsigned long long)(size_t)srow + (unsigned)(lane * 16);
    unsigned long long go = (unsigned long long)(size_t)orow + (unsigned)(lane * 16);

    // Async DMA: global -> LDS (ASYNCcnt), wait, LDS -> global.
    // INST_OFFSET applies to both the LDS and global sides (ISA 08 §4.4).
    asm volatile(
        "global_load_async_to_lds_b128 %0, %1, off\n\t"
        "global_load_async_to_lds_b128 %0, %1, off offset:512\n\t"
        "global_load_async_to_lds_b128 %0, %1, off offset:1024\n\t"
        "s_wait_asynccnt 0"
        :: "v"(lds_a), "v"(ga) : "memory");

    asm volatile(
        "global_store_async_from_lds_b128 %1, %0, off\n\t"
        "global_store_async_from_lds_b128 %1, %0, off offset:512\n\t"
        "global_store_async_from_lds_b128 %1, %0, off offset:1024"
        :: "v"(lds_a), "v"(go) : "memory");
    // s_endpgm performs an implicit wait-idle, draining the async stores.
}

extern "C" void kernel_launch(void* const* d_in, const int* in_sizes, int n_in,
                              void* d_out, int out_size, void* d_ws, size_t ws_size,
                              hipStream_t stream) {
    const float* x   = (const float*)d_in[0];   // (32, 1024, 384) f32
    const int*   dur = (const int*)d_in[1];     // (32, 1024) i32
    float*       out = (float*)d_out;           // (32, 4096, 384) f32
    int*         cum = (int*)d_ws;              // 32*1024 i32 = 128KB scratch

    lr_scan_kernel<<<dim3(BATCH), dim3(256), 0, stream>>>(dur, cum);
    lr_gather_kernel<<<dim3(MAX_FRAMES / WAVES_PER_BLOCK, BATCH), dim3(256), 0, stream>>>(x, cum, out);
}